// SMPL_layer_59974923322038
// MI455X (gfx1250) — compile-verified
//
#include <hip/hip_runtime.h>
#include <hip/hip_bf16.h>
#include <math.h>

// ---------------- problem constants ----------------
#define BB   512          // batch
#define VV   6890         // vertices
#define JJ   24           // joints
#define NB   10           // betas
#define PP   207          // pose features (J-1)*9
#define KP   208          // padded K for WMMA (pose_feature col 207 == 0)
#define NN   (VV*3)       // 20670, pose-blend GEMM N / row length of v_shaped
#define H36  17

typedef __attribute__((ext_vector_type(2))) float v2f;
typedef __attribute__((ext_vector_type(8))) float v8f;

// ---------------- 1: Rodrigues + pose_feature ----------------
__global__ void k_rodrigues(const float* __restrict__ pose_aa,
                            const float* __restrict__ gorient,
                            float* __restrict__ rotm,      // out region 3: (B,24,9)
                            float* __restrict__ pf) {      // ws: (B,208)
    int idx = blockIdx.x * blockDim.x + threadIdx.x;
    if (idx >= BB * JJ) return;
    int b = idx / JJ, j = idx % JJ;
    float x, y, z;
    if (j == 0) { x = gorient[b*3+0]; y = gorient[b*3+1]; z = gorient[b*3+2]; }
    else { int o = b*69 + (j-1)*3; x = pose_aa[o]; y = pose_aa[o+1]; z = pose_aa[o+2]; }
    float xe = x + 1e-8f, ye = y + 1e-8f, ze = z + 1e-8f;
    float angle = sqrtf(xe*xe + ye*ye + ze*ze);
    float inv = 1.0f / angle;
    float kx = x*inv, ky = y*inv, kz = z*inv;
    float s = sinf(angle), c = cosf(angle), t = 1.0f - c;
    float R[9];
    R[0] = 1.f - t*(ky*ky + kz*kz); R[1] = -s*kz + t*kx*ky;          R[2] =  s*ky + t*kx*kz;
    R[3] =  s*kz + t*kx*ky;         R[4] = 1.f - t*(kx*kx + kz*kz);  R[5] = -s*kx + t*ky*kz;
    R[6] = -s*ky + t*kx*kz;         R[7] =  s*kx + t*ky*kz;          R[8] = 1.f - t*(kx*kx + ky*ky);
    float* ro = rotm + (b*JJ + j)*9;
    #pragma unroll
    for (int i = 0; i < 9; ++i) ro[i] = R[i];
    if (j > 0) {
        float* po = pf + b*KP + (j-1)*9;
        #pragma unroll
        for (int i = 0; i < 9; ++i) po[i] = R[i] - ((i == 0 || i == 4 || i == 8) ? 1.f : 0.f);
    } else {
        pf[b*KP + PP] = 0.f;   // K padding column
    }
}

// ---------------- 2: v_shaped = v_template + shapedirs @ betas ----------------
__global__ void k_vshaped(const float* __restrict__ betas,
                          const float* __restrict__ shapedirs,   // (V,3,NB)
                          const float* __restrict__ v_template,  // (V,3)
                          float* __restrict__ vshaped) {         // ws: (B, NN)
    int idx = blockIdx.x * blockDim.x + threadIdx.x;
    if (idx >= BB * VV) return;
    int b = idx / VV, v = idx % VV;
    float bb[NB];
    #pragma unroll
    for (int l = 0; l < NB; ++l) bb[l] = betas[b*NB + l];
    const float* sd = shapedirs + v*3*NB;
    #pragma unroll
    for (int d = 0; d < 3; ++d) {
        float acc = v_template[v*3 + d];
        #pragma unroll
        for (int l = 0; l < NB; ++l) acc += bb[l] * sd[d*NB + l];
        vshaped[b*NN + v*3 + d] = acc;
    }
}

// ---------------- 3: joints_rest = J_regressor @ v_shaped ----------------
__global__ void k_joints(const float* __restrict__ Jreg,     // (J,V)
                         const float* __restrict__ vshaped,  // (B,NN)
                         float* __restrict__ jrest) {        // ws: (B,J,3)
    int j = blockIdx.x, b = blockIdx.y, tid = threadIdx.x;
    __shared__ float sred[3 * 256];
    const float* row = Jreg + (size_t)j * VV;
    const float* vs  = vshaped + (size_t)b * NN;
    float a0 = 0.f, a1 = 0.f, a2 = 0.f;
    for (int v = tid; v < VV; v += 256) {
        float w = row[v];
        a0 += w * vs[v*3+0]; a1 += w * vs[v*3+1]; a2 += w * vs[v*3+2];
    }
    sred[tid] = a0; sred[256 + tid] = a1; sred[512 + tid] = a2;
    __syncthreads();
    for (int s = 128; s > 0; s >>= 1) {
        if (tid < s) {
            sred[tid]       += sred[tid + s];
            sred[256 + tid] += sred[256 + tid + s];
            sred[512 + tid] += sred[512 + tid + s];
        }
        __syncthreads();
    }
    if (tid == 0) {
        float* o = jrest + (b*JJ + j)*3;
        o[0] = sred[0]; o[1] = sred[256]; o[2] = sred[512];
    }
}

// ---------------- 4: kinematic chain -> posed_joints, rel_A ----------------
// constexpr parent table + full unroll => all indices compile-time, jr[] stays
// in VGPRs (no scratch). A chain staged through ws (same-thread order is kept).
__global__ void k_chain(const float* __restrict__ rotm,     // (B,J,9)
                        const float* __restrict__ jrest,    // (B,J,3)
                        const float* __restrict__ transl,   // (B,3)
                        float* __restrict__ Achain,         // ws: (B,J,12)
                        float* __restrict__ relA,           // ws: (B,J,12)
                        float* __restrict__ pjoints) {      // out region 2: (B,J,3)
    constexpr int parents[JJ] =
        {-1, 0, 0, 0, 1, 2, 3, 4, 5, 6, 7, 8, 9, 9, 9, 12, 13, 14, 16, 17, 18, 19, 20, 21};
    int b = blockIdx.x * blockDim.x + threadIdx.x;
    if (b >= BB) return;
    float jr[JJ][3];
    #pragma unroll
    for (int j = 0; j < JJ; ++j) {
        jr[j][0] = jrest[(b*JJ + j)*3 + 0];
        jr[j][1] = jrest[(b*JJ + j)*3 + 1];
        jr[j][2] = jrest[(b*JJ + j)*3 + 2];
    }
    float tx = transl[b*3+0], ty = transl[b*3+1], tz = transl[b*3+2];
    #pragma unroll
    for (int j = 0; j < JJ; ++j) {
        const float* R = rotm + (b*JJ + j)*9;
        float Aj[12];
        if (j == 0) {
            #pragma unroll
            for (int r = 0; r < 3; ++r) {
                Aj[r*4+0] = R[r*3+0]; Aj[r*4+1] = R[r*3+1]; Aj[r*4+2] = R[r*3+2];
            }
            Aj[3] = jr[0][0]; Aj[7] = jr[0][1]; Aj[11] = jr[0][2];
        } else {
            constexpr int p0 = 0;                 // placate indexing; real p below
            const int p = parents[j] < 0 ? p0 : parents[j];
            const float r0 = jr[j][0] - jr[p][0];
            const float r1 = jr[j][1] - jr[p][1];
            const float r2 = jr[j][2] - jr[p][2];
            const float* Ap = Achain + (b*JJ + p)*12;
            float ap[12];
            #pragma unroll
            for (int i = 0; i < 12; ++i) ap[i] = Ap[i];
            #pragma unroll
            for (int r = 0; r < 3; ++r) {
                #pragma unroll
                for (int c = 0; c < 3; ++c)
                    Aj[r*4+c] = ap[r*4+0]*R[0*3+c] + ap[r*4+1]*R[1*3+c] + ap[r*4+2]*R[2*3+c];
                Aj[r*4+3] = ap[r*4+0]*r0 + ap[r*4+1]*r1 + ap[r*4+2]*r2 + ap[r*4+3];
            }
        }
        float* Ao = Achain + (b*JJ + j)*12;
        #pragma unroll
        for (int i = 0; i < 12; ++i) Ao[i] = Aj[i];
        pjoints[(b*JJ + j)*3 + 0] = Aj[3]  + tx;
        pjoints[(b*JJ + j)*3 + 1] = Aj[7]  + ty;
        pjoints[(b*JJ + j)*3 + 2] = Aj[11] + tz;
        float* Ro = relA + (b*JJ + j)*12;
        #pragma unroll
        for (int r = 0; r < 3; ++r) {
            float corr = Aj[r*4+0]*jr[j][0] + Aj[r*4+1]*jr[j][1] + Aj[r*4+2]*jr[j][2];
            Ro[r*4+0] = Aj[r*4+0]; Ro[r*4+1] = Aj[r*4+1]; Ro[r*4+2] = Aj[r*4+2];
            Ro[r*4+3] = Aj[r*4+3] - corr;
        }
    }
}

// ---------------- 5: WMMA pose-blend GEMM: v_posed = v_shaped + pf @ posedirs ----------------
// M=512 (batch), K=208 (padded), N=20670.  f32 WMMA 16x16x4 keeps full precision.
// Block: 8 waves; tile 16(M) x 128(N); pose_feature tile staged in LDS.
__global__ void k_poseblend(const float* __restrict__ pf,        // (B,KP), col 207 zeroed
                            const float* __restrict__ posedirs,  // (207, NN)
                            const float* __restrict__ vshaped,   // (B, NN)
                            float* __restrict__ vposed) {        // ws: (B, NN)
    __shared__ float s_pf[16 * KP];
    const int tid  = threadIdx.x;
    const int m0   = blockIdx.y * 16;
    const int wave = tid >> 5;
    const int lane = tid & 31;
    for (int i = tid; i < 16 * KP; i += 256) {
        int r = i / KP, k = i % KP;
        s_pf[i] = pf[(m0 + r)*KP + k];
    }
    __syncthreads();
    const int n0   = blockIdx.x * 128 + wave * 16;
    const int half = lane >> 4;        // 0: lanes 0-15, 1: lanes 16-31
    const int l    = lane & 15;
    const int col  = n0 + l;
    const int colc = (col < NN) ? col : (NN - 1);   // clamped load (guarded store below)
    v8f acc = {};
    #pragma unroll 4
    for (int kk = 0; kk < KP/4; ++kk) {
        const int k0 = kk*4 + 2*half;
        v2f a, bf;
        a.x = s_pf[l*KP + k0];
        a.y = s_pf[l*KP + k0 + 1];
        // clamp B row to 206: the clamped rows multiply the zero pad column of A
        const int r0 = (k0     < PP) ? k0     : (PP - 1);
        const int r1 = (k0 + 1 < PP) ? k0 + 1 : (PP - 1);
        bf.x = posedirs[(size_t)r0*NN + colc];
        bf.y = posedirs[(size_t)r1*NN + colc];
        acc = __builtin_amdgcn_wmma_f32_16x16x4_f32(
                  /*neg_a=*/false, a, /*neg_b=*/false, bf,
                  /*c_mod=*/(short)0, acc, /*reuse_a=*/false, /*reuse_b=*/false);
    }
    if (col < NN) {
        #pragma unroll
        for (int r = 0; r < 8; ++r) {
            const int m = m0 + r + 8*half;     // C/D layout: VGPR r = row r (lanes 0-15), r+8 (16-31)
            vposed[(size_t)m*NN + col] = acc[r] + vshaped[(size_t)m*NN + col];
        }
    }
}

// ---------------- 6: LBS skinning -> verts (+transl) ----------------
__global__ void k_skin(const float* __restrict__ lbs,      // (V,J)
                       const float* __restrict__ relA,     // (B,J,12)
                       const float* __restrict__ vposed,   // (B,NN)
                       const float* __restrict__ transl,   // (B,3)
                       float* __restrict__ verts) {        // out region 1: (B,V,3)
    const int b = blockIdx.y;
    __shared__ float sA[JJ * 12];
    for (int i = threadIdx.x; i < JJ*12; i += 256) sA[i] = relA[b*JJ*12 + i];
    __syncthreads();
    const int v = blockIdx.x * 256 + threadIdx.x;
    if (v >= VV) return;
    float T[12];
    #pragma unroll
    for (int i = 0; i < 12; ++i) T[i] = 0.f;
    const float* w = lbs + (size_t)v * JJ;
    #pragma unroll
    for (int j = 0; j < JJ; ++j) {
        const float wj = w[j];
        #pragma unroll
        for (int i = 0; i < 12; ++i) T[i] += wj * sA[j*12 + i];
    }
    const float px = vposed[(size_t)b*NN + v*3 + 0];
    const float py = vposed[(size_t)b*NN + v*3 + 1];
    const float pz = vposed[(size_t)b*NN + v*3 + 2];
    const float ox = T[0]*px + T[1]*py + T[2]*pz  + T[3];
    const float oy = T[4]*px + T[5]*py + T[6]*pz  + T[7];
    const float oz = T[8]*px + T[9]*py + T[10]*pz + T[11];
    verts[(size_t)b*NN + v*3 + 0] = ox + transl[b*3+0];
    verts[(size_t)b*NN + v*3 + 1] = oy + transl[b*3+1];
    verts[(size_t)b*NN + v*3 + 2] = oz + transl[b*3+2];
}

// ---------------- 7: joints_from_verts = J_regressor_h36m @ (verts+t) ----------------
// rows of J_regressor_h36m sum to 1, so regressing verts+t directly yields jfv+t.
__global__ void k_jverts(const float* __restrict__ Jh,     // (17,V)
                         const float* __restrict__ verts,  // (B,V,3) already + t
                         float* __restrict__ jfv) {        // out region 4: (B,17,3)
    int h = blockIdx.x, b = blockIdx.y, tid = threadIdx.x;
    __shared__ float sred[3 * 256];
    const float* row = Jh + (size_t)h * VV;
    const float* vs  = verts + (size_t)b * NN;
    float a0 = 0.f, a1 = 0.f, a2 = 0.f;
    for (int v = tid; v < VV; v += 256) {
        float w = row[v];
        a0 += w * vs[v*3+0]; a1 += w * vs[v*3+1]; a2 += w * vs[v*3+2];
    }
    sred[tid] = a0; sred[256 + tid] = a1; sred[512 + tid] = a2;
    __syncthreads();
    for (int s = 128; s > 0; s >>= 1) {
        if (tid < s) {
            sred[tid]       += sred[tid + s];
            sred[256 + tid] += sred[256 + tid + s];
            sred[512 + tid] += sred[512 + tid + s];
        }
        __syncthreads();
    }
    if (tid == 0) {
        float* o = jfv + (b*H36 + h)*3;
        o[0] = sred[0]; o[1] = sred[256]; o[2] = sred[512];
    }
}

// ---------------- launch ----------------
extern "C" void kernel_launch(void* const* d_in, const int* in_sizes, int n_in,
                              void* d_out, int out_size, void* d_ws, size_t ws_size,
                              hipStream_t stream) {
    const float* pose_aa  = (const float*)d_in[0];   // (B,69)
    const float* betas    = (const float*)d_in[1];   // (B,10)
    const float* gorient  = (const float*)d_in[2];   // (B,3)
    const float* transl   = (const float*)d_in[3];   // (B,3)
    const float* v_templ  = (const float*)d_in[4];   // (V,3)
    const float* shapedir = (const float*)d_in[5];   // (V,3,10)
    const float* posedirs = (const float*)d_in[6];   // (207, V*3)
    const float* Jreg     = (const float*)d_in[7];   // (24,V)
    const float* Jh36     = (const float*)d_in[8];   // (17,V)
    const float* lbs      = (const float*)d_in[9];   // (V,24)

    float* out      = (float*)d_out;
    float* verts    = out;                                   // B*V*3
    float* pjoints  = verts   + (size_t)BB*VV*3;             // B*24*3
    float* rotm     = pjoints + (size_t)BB*JJ*3;             // B*24*9
    float* jfv      = rotm    + (size_t)BB*JJ*9;             // B*17*3

    float* W        = (float*)d_ws;
    float* pf       = W;                                     // B*208
    float* vshaped  = pf      + (size_t)BB*KP;               // B*NN
    float* vposed   = vshaped + (size_t)BB*NN;               // B*NN
    float* jrest    = vposed  + (size_t)BB*NN;               // B*J*3
    float* Achain   = jrest   + (size_t)BB*JJ*3;             // B*J*12
    float* relA     = Achain  + (size_t)BB*JJ*12;            // B*J*12

    (void)in_sizes; (void)n_in; (void)out_size; (void)ws_size;

    k_rodrigues<<<dim3((BB*JJ + 255)/256), dim3(256), 0, stream>>>(pose_aa, gorient, rotm, pf);
    k_vshaped  <<<dim3((BB*VV + 255)/256), dim3(256), 0, stream>>>(betas, shapedir, v_templ, vshaped);
    k_joints   <<<dim3(JJ, BB),            dim3(256), 0, stream>>>(Jreg, vshaped, jrest);
    k_chain    <<<dim3((BB + 63)/64),      dim3(64),  0, stream>>>(rotm, jrest, transl, Achain, relA, pjoints);
    k_poseblend<<<dim3((NN + 127)/128, BB/16), dim3(256), 0, stream>>>(pf, posedirs, vshaped, vposed);
    k_skin     <<<dim3((VV + 255)/256, BB), dim3(256), 0, stream>>>(lbs, relA, vposed, transl, verts);
    k_jverts   <<<dim3(H36, BB),           dim3(256), 0, stream>>>(Jh36, verts, jfv);
}